// HAN_37443524886887
// MI455X (gfx1250) — compile-verified
//
#include <hip/hip_runtime.h>
#include <cmath>

// ---------------------------------------------------------------------------
// HAN layer for MI455X (gfx1250, wave32).
//  - f32 WMMA (V_WMMA_F32_16X16X4_F32) for all dense GEMMs, K templated so the
//    whole WMMA chain fully unrolls (no loop SALU, immediate-offset loads).
//  - A-tiles staged HBM -> LDS via GLOBAL_LOAD_ASYNC_TO_LDS_B128 (ASYNCcnt).
//  - Edge phase: 3 bandwidth-optimal passes with float atomics.
// ---------------------------------------------------------------------------

typedef float v2f __attribute__((ext_vector_type(2)));
typedef float v8f __attribute__((ext_vector_type(8)));

#define HID 128   // H*D
#define NH  8     // heads
#define HD  16    // head dim

// ---------------- utility ----------------

__global__ void fill_f32(float* __restrict__ p, float v, long long n) {
  long long i = (long long)blockIdx.x * blockDim.x + threadIdx.x;
  if (i < n) p[i] = v;
}

__device__ inline void atomicMaxF(float* addr, float val) {
  unsigned int* ua = reinterpret_cast<unsigned int*>(addr);
  unsigned int cur = *ua;
  while (__uint_as_float(cur) < val) {
    unsigned int prev = atomicCAS(ua, cur, __float_as_uint(val));
    if (prev == cur) break;
    cur = prev;
  }
}

// Async HBM->LDS stage of a 16 x K f32 tile (row-major, contiguous in LDS).
// Each thread moves 16-byte chunks; ASYNCcnt tracked, barrier after wait.
template <int K>
__device__ inline void stage_tile_async(const float* __restrict__ X,
                                        float* As, int m0, int N, int tid) {
  const unsigned lds_base = (unsigned)(unsigned long long)(void*)As;
  constexpr int CPR = K >> 2;                  // 16B chunks per row
  #pragma unroll
  for (int c0 = 0; c0 < 4 * K; c0 += 256) {    // 4*K chunks cover 16 rows
    int c   = c0 + tid;
    int row = c / CPR;
    int cc  = c - row * CPR;
    int rr  = m0 + row; if (rr >= N) rr = N - 1;
    unsigned long long gaddr =
        (unsigned long long)(const void*)(X + (size_t)rr * K) +
        (unsigned long long)cc * 16ull;
    unsigned laddr = lds_base + (unsigned)c * 16u;
    asm volatile("global_load_async_to_lds_b128 %0, %1, off"
                 :: "v"(laddr), "v"(gaddr) : "memory");
  }
  asm volatile("s_wait_asynccnt 0x0" ::: "memory");
  __syncthreads();
}

// ---------------- dense projection: Hout[N,128] = X[N,K] @ W[K,128] + bias ---
// One wave -> one 16x16 output tile via V_WMMA_F32_16X16X4_F32; K fully
// unrolled (K/4 chained WMMAs). 8 waves/block cover all 128 output columns.
template <int K>
__global__ __launch_bounds__(256) void proj_wmma(
    const float* __restrict__ X, const float* __restrict__ W,
    const float* __restrict__ bias, float* __restrict__ Hout, int N)
{
  __shared__ float As[16 * K];
  const int m0  = blockIdx.x * 16;
  const int tid = threadIdx.x;

  stage_tile_async<K>(X, As, m0, N, tid);

  const int lane = tid & 31;
  const int half = lane >> 4;        // lanes 16-31 hold K+2 / M+8 halves
  const int mod  = lane & 15;
  const int n0   = (tid >> 5) * 16;  // wave id * 16
  const int col  = n0 + mod;

  v8f acc = {0.f, 0.f, 0.f, 0.f, 0.f, 0.f, 0.f, 0.f};

  #pragma unroll
  for (int k0 = 0; k0 < K; k0 += 4) {
    const int ka = k0 + 2 * half;
    v2f a, b;
    a.x = As[mod * K + ka];
    a.y = As[mod * K + ka + 1];
    b.x = W[(size_t)ka       * HID + col];
    b.y = W[(size_t)(ka + 1) * HID + col];
    acc = __builtin_amdgcn_wmma_f32_16x16x4_f32(
        false, a, false, b, (short)0, acc, false, false);
  }

  const float bv = bias[col];
  #pragma unroll
  for (int r = 0; r < 8; ++r) {
    int row = m0 + r + 8 * half;
    if (row < N) Hout[(size_t)row * HID + col] = acc[r] + bv;
  }
}

// ------------- semantic attention GEMM: wsum[t,128] += sum_M tanh(relu(O)@Wk+bk)
__global__ __launch_bounds__(256) void semantic_wmma(
    const float* __restrict__ o_pa, const float* __restrict__ o_da,
    const float* __restrict__ Wk, const float* __restrict__ bk,
    float* __restrict__ wsum, int N)
{
  const float* O = (blockIdx.y == 0) ? o_pa : o_da;
  __shared__ float As[16 * HID];
  const int m0  = blockIdx.x * 16;
  const int tid = threadIdx.x;

  stage_tile_async<HID>(O, As, m0, N, tid);

  const int lane = tid & 31;
  const int half = lane >> 4;
  const int mod  = lane & 15;
  const int n0   = (tid >> 5) * 16;
  const int col  = n0 + mod;

  v8f acc = {0.f, 0.f, 0.f, 0.f, 0.f, 0.f, 0.f, 0.f};

  #pragma unroll
  for (int k0 = 0; k0 < HID; k0 += 4) {
    const int ka = k0 + 2 * half;
    v2f a, b;
    a.x = fmaxf(As[mod * HID + ka], 0.f);       // fused relu on LDS read
    a.y = fmaxf(As[mod * HID + ka + 1], 0.f);
    b.x = Wk[(size_t)ka       * HID + col];
    b.y = Wk[(size_t)(ka + 1) * HID + col];
    acc = __builtin_amdgcn_wmma_f32_16x16x4_f32(
        false, a, false, b, (short)0, acc, false, false);
  }

  const float bkv = bk[col];
  float partial = 0.f;
  #pragma unroll
  for (int r = 0; r < 8; ++r) {
    int row = m0 + r + 8 * half;
    if (row < N) partial += tanhf(acc[r] + bkv);
  }
  atomicAdd(&wsum[blockIdx.y * HID + col], partial);
}

// ---------------- per-node attention scores: out[n,h] = sum_d h[n,h,d]*att[h,d]
__global__ void att_scores(const float* __restrict__ Hm,
                           const float* __restrict__ att,
                           float* __restrict__ out, int N)
{
  int idx = blockIdx.x * blockDim.x + threadIdx.x;
  if (idx >= N * NH) return;
  int h = idx & (NH - 1);
  int node = idx >> 3;
  const float* row = Hm + (size_t)node * HID + h * HD;
  const float* av  = att + h * HD;
  float s = 0.f;
  #pragma unroll
  for (int d = 0; d < HD; ++d) s += row[d] * av[d];
  out[idx] = s;
}

// ---------------- edge pass 1: segment max of leaky-relu logits --------------
__global__ void edge_max(const int* __restrict__ src, const int* __restrict__ dst,
                         const float* __restrict__ as, const float* __restrict__ ad,
                         float* __restrict__ m, long long E)
{
  long long idx = (long long)blockIdx.x * blockDim.x + threadIdx.x;
  if (idx >= E * NH) return;
  int h = (int)(idx & (NH - 1));
  long long e = idx >> 3;
  int sn = src[e], dn = dst[e];
  float a = as[(size_t)sn * NH + h] + ad[(size_t)dn * NH + h];
  a = (a >= 0.f) ? a : 0.2f * a;                  // leaky_relu(0.2)
  atomicMaxF(&m[(size_t)dn * NH + h], a);
}

// ---------------- edge pass 2: exp + segment sum -----------------------------
__global__ void edge_exp(const int* __restrict__ src, const int* __restrict__ dst,
                         const float* __restrict__ as, const float* __restrict__ ad,
                         const float* __restrict__ m, float* __restrict__ ebuf,
                         float* __restrict__ ssum, long long E)
{
  long long idx = (long long)blockIdx.x * blockDim.x + threadIdx.x;
  if (idx >= E * NH) return;
  int h = (int)(idx & (NH - 1));
  long long e = idx >> 3;
  int sn = src[e], dn = dst[e];
  float a = as[(size_t)sn * NH + h] + ad[(size_t)dn * NH + h];
  a = (a >= 0.f) ? a : 0.2f * a;
  float ev = __expf(a - m[(size_t)dn * NH + h]);
  ebuf[idx] = ev;
  atomicAdd(&ssum[(size_t)dn * NH + h], ev);
}

// ---------------- edge pass 3: normalized message scatter-add ----------------
// 1 wave per edge, each lane owns 4 contiguous floats (same head).
__global__ void edge_scatter(const int* __restrict__ src, const int* __restrict__ dst,
                             const float* __restrict__ Hsrc,
                             const float* __restrict__ ebuf,
                             const float* __restrict__ ssum,
                             float* __restrict__ o, long long E)
{
  long long gid = (long long)blockIdx.x * blockDim.x + threadIdx.x;
  int lane = (int)(gid & 31);
  long long e = gid >> 5;
  if (e >= E) return;
  int sn = src[e], dn = dst[e];
  int h = lane >> 2;                               // 4 floats per lane, 16/head
  float w = ebuf[e * NH + h] / (ssum[(size_t)dn * NH + h] + 1e-16f);
  const float4 v = *(const float4*)(Hsrc + (size_t)sn * HID + lane * 4);
  float* ob = o + (size_t)dn * HID + lane * 4;
  atomicAdd(ob + 0, v.x * w);
  atomicAdd(ob + 1, v.y * w);
  atomicAdd(ob + 2, v.z * w);
  atomicAdd(ob + 3, v.w * w);
}

// ---------------- semantic softmax over 2 metapaths --------------------------
__global__ void beta_kernel(const float* __restrict__ wsum,
                            const float* __restrict__ q,
                            float* __restrict__ beta, float invNa)
{
  float b0 = 0.f, b1 = 0.f;
  for (int j = 0; j < HID; ++j) {
    b0 += wsum[j]        * invNa * q[j];
    b1 += wsum[HID + j]  * invNa * q[j];
  }
  float mx = fmaxf(b0, b1);
  float e0 = __expf(b0 - mx), e1 = __expf(b1 - mx);
  float inv = 1.f / (e0 + e1);
  beta[0] = e0 * inv;
  beta[1] = e1 * inv;
}

// ---------------- final combine + classifier + softmax -----------------------
__global__ void final_kernel(const float* __restrict__ o_pa,
                             const float* __restrict__ o_da,
                             const float* __restrict__ beta,
                             const float* __restrict__ Wl,
                             const float* __restrict__ bl,
                             float* __restrict__ out, int Na)
{
  int n = blockIdx.x * blockDim.x + threadIdx.x;
  if (n >= Na) return;
  float b0 = beta[0], b1 = beta[1];
  float l0 = bl[0],  l1 = bl[1];
  const float* pa = o_pa + (size_t)n * HID;
  const float* da = o_da + (size_t)n * HID;
  #pragma unroll 4
  for (int j = 0; j < HID; ++j) {
    float s = b0 * fmaxf(pa[j], 0.f) + b1 * fmaxf(da[j], 0.f);
    l0 += s * Wl[j * 2 + 0];
    l1 += s * Wl[j * 2 + 1];
  }
  float mx = fmaxf(l0, l1);
  float e0 = __expf(l0 - mx), e1 = __expf(l1 - mx);
  float inv = 1.f / (e0 + e1);
  out[n * 2 + 0] = e0 * inv;
  out[n * 2 + 1] = e1 * inv;
}

// ---------------------------------------------------------------------------

extern "C" void kernel_launch(void* const* d_in, const int* in_sizes, int n_in,
                              void* d_out, int out_size, void* d_ws, size_t ws_size,
                              hipStream_t stream) {
  const float* x_adm  = (const float*)d_in[0];
  const float* x_pat  = (const float*)d_in[1];
  const float* x_diag = (const float*)d_in[2];
  const int* e_pa_src = (const int*)d_in[3];
  const int* e_pa_dst = (const int*)d_in[4];
  const int* e_da_src = (const int*)d_in[5];
  const int* e_da_dst = (const int*)d_in[6];
  const float* Wa = (const float*)d_in[7];
  const float* ba = (const float*)d_in[8];
  const float* Wp = (const float*)d_in[9];
  const float* bp = (const float*)d_in[10];
  const float* Wd = (const float*)d_in[11];
  const float* bd = (const float*)d_in[12];
  const float* ats_pa = (const float*)d_in[13];
  const float* atd_pa = (const float*)d_in[14];
  const float* ats_da = (const float*)d_in[15];
  const float* atd_da = (const float*)d_in[16];
  const float* Wk = (const float*)d_in[17];
  const float* bk = (const float*)d_in[18];
  const float* q  = (const float*)d_in[19];
  const float* Wl = (const float*)d_in[20];
  const float* bl = (const float*)d_in[21];
  float* out = (float*)d_out;

  const int Na = in_sizes[0] / 128;
  const int Np = in_sizes[1] / 64;
  const int Nd = in_sizes[2] / 64;
  const long long E_pa = in_sizes[3];
  const long long E_da = in_sizes[5];

  // ---- workspace carve (floats) ----
  float* w = (float*)d_ws;
  size_t off = 0;
  float* h_a  = w + off; off += (size_t)Na * HID;
  float* h_p  = w + off; off += (size_t)Np * HID;
  float* h_d  = w + off; off += (size_t)Nd * HID;
  float* asp  = w + off; off += (size_t)Np * NH;   // a_s (pat)   for p->a
  float* adp  = w + off; off += (size_t)Na * NH;   // a_d (adm)   for p->a
  float* asd  = w + off; off += (size_t)Nd * NH;   // a_s (diag)  for d->a
  float* add_ = w + off; off += (size_t)Na * NH;   // a_d (adm)   for d->a
  float* m_pa = w + off; off += (size_t)Na * NH;
  float* s_pa = w + off; off += (size_t)Na * NH;
  float* m_da = w + off; off += (size_t)Na * NH;
  float* s_da = w + off; off += (size_t)Na * NH;
  float* eb_pa = w + off; off += (size_t)E_pa * NH;
  float* eb_da = w + off; off += (size_t)E_da * NH;
  float* o_pa = w + off; off += (size_t)Na * HID;
  float* o_da = w + off; off += (size_t)Na * HID;
  float* wsum = w + off; off += 2 * HID;
  float* beta = w + off; off += 2;

  const dim3 blk(256);
  auto gr1 = [](long long n) { return dim3((unsigned)((n + 255) / 256)); };

  // ---- init accumulators (every call: graph-replay safe) ----
  fill_f32<<<gr1((long long)Na * HID), blk, 0, stream>>>(o_pa, 0.f, (long long)Na * HID);
  fill_f32<<<gr1((long long)Na * HID), blk, 0, stream>>>(o_da, 0.f, (long long)Na * HID);
  fill_f32<<<gr1((long long)Na * NH), blk, 0, stream>>>(s_pa, 0.f, (long long)Na * NH);
  fill_f32<<<gr1((long long)Na * NH), blk, 0, stream>>>(s_da, 0.f, (long long)Na * NH);
  fill_f32<<<gr1((long long)Na * NH), blk, 0, stream>>>(m_pa, -INFINITY, (long long)Na * NH);
  fill_f32<<<gr1((long long)Na * NH), blk, 0, stream>>>(m_da, -INFINITY, (long long)Na * NH);
  fill_f32<<<gr1(2 * HID), blk, 0, stream>>>(wsum, 0.f, 2 * HID);

  // ---- dense projections (WMMA + async LDS staging, K fully unrolled) ----
  proj_wmma<128><<<dim3((Na + 15) / 16), blk, 0, stream>>>(x_adm, Wa, ba, h_a, Na);
  proj_wmma< 64><<<dim3((Np + 15) / 16), blk, 0, stream>>>(x_pat, Wp, bp, h_p, Np);
  proj_wmma< 64><<<dim3((Nd + 15) / 16), blk, 0, stream>>>(x_diag, Wd, bd, h_d, Nd);

  // ---- per-node attention scores ----
  att_scores<<<gr1((long long)Np * NH), blk, 0, stream>>>(h_p, ats_pa, asp, Np);
  att_scores<<<gr1((long long)Na * NH), blk, 0, stream>>>(h_a, atd_pa, adp, Na);
  att_scores<<<gr1((long long)Nd * NH), blk, 0, stream>>>(h_d, ats_da, asd, Nd);
  att_scores<<<gr1((long long)Na * NH), blk, 0, stream>>>(h_a, atd_da, add_, Na);

  // ---- edge-type p->a ----
  edge_max<<<gr1(E_pa * NH), blk, 0, stream>>>(e_pa_src, e_pa_dst, asp, adp, m_pa, E_pa);
  edge_exp<<<gr1(E_pa * NH), blk, 0, stream>>>(e_pa_src, e_pa_dst, asp, adp, m_pa, eb_pa, s_pa, E_pa);
  edge_scatter<<<gr1(E_pa * 32), blk, 0, stream>>>(e_pa_src, e_pa_dst, h_p, eb_pa, s_pa, o_pa, E_pa);

  // ---- edge-type d->a ----
  edge_max<<<gr1(E_da * NH), blk, 0, stream>>>(e_da_src, e_da_dst, asd, add_, m_da, E_da);
  edge_exp<<<gr1(E_da * NH), blk, 0, stream>>>(e_da_src, e_da_dst, asd, add_, m_da, eb_da, s_da, E_da);
  edge_scatter<<<gr1(E_da * 32), blk, 0, stream>>>(e_da_src, e_da_dst, h_d, eb_da, s_da, o_da, E_da);

  // ---- semantic attention (WMMA + fused relu/tanh/mean reduction) ----
  semantic_wmma<<<dim3((Na + 15) / 16, 2), blk, 0, stream>>>(o_pa, o_da, Wk, bk, wsum, Na);
  beta_kernel<<<dim3(1), dim3(1), 0, stream>>>(wsum, q, beta, 1.f / (float)Na);

  // ---- final combine + classifier ----
  final_kernel<<<gr1(Na), blk, 0, stream>>>(o_pa, o_da, beta, Wl, bl, out, Na);
}